// Damping_layer_67336497266742
// MI455X (gfx1250) — compile-verified
//
#include <hip/hip_runtime.h>

// out = kipf_term - lbda[row] * input_term   (100000 x 256, f32)
// Pure streaming elementwise op: HBM-bound (~307 MB traffic, ~13 us at 23.3 TB/s).
// Strategy: b128 vector memory ops + non-temporal hints (footprint > 192 MB L2),
// cached scalar-ish load for the small reused lbda vector, wave32-friendly 256-thread blocks.

typedef float v4f __attribute__((ext_vector_type(4)));

__global__ __launch_bounds__(256) void damping_layer_kernel(
    const v4f* __restrict__ input4,   // input_term viewed as float4
    const v4f* __restrict__ kipf4,    // kipf_term viewed as float4
    const float* __restrict__ lbda,   // per-row damping factor
    v4f* __restrict__ out4,
    int n4)                            // total float4 count = N_NODES * N_FEATURES / 4
{
    int i = blockIdx.x * blockDim.x + threadIdx.x;
    if (i >= n4) return;

    // 256 features / 4 = 64 float4s per row -> row index = i >> 6.
    // Reused 64x across lanes; keep it in cache (regular load, no NT hint).
    float l = lbda[i >> 6];

    // Streamed-once operands: non-temporal b128 loads so they don't thrash L2.
    v4f a = __builtin_nontemporal_load(&input4[i]);
    v4f k = __builtin_nontemporal_load(&kipf4[i]);

    // k - l*a -> v_fma_f32 with negated scalar per component.
    v4f r = k - l * a;

    // Streamed-once result: non-temporal b128 store.
    __builtin_nontemporal_store(r, &out4[i]);
}

extern "C" void kernel_launch(void* const* d_in, const int* in_sizes, int n_in,
                              void* d_out, int out_size, void* d_ws, size_t ws_size,
                              hipStream_t stream) {
    const v4f*   input4 = (const v4f*)d_in[0];    // input_term: 100000*256 f32
    const v4f*   kipf4  = (const v4f*)d_in[1];    // kipf_term:  100000*256 f32
    const float* lbda   = (const float*)d_in[2];  // lbda: 100000 f32
    // d_in[3] = spar (scalar, ==1 in the reference path) -> unused

    v4f* out4 = (v4f*)d_out;

    int n4 = in_sizes[0] / 4;                     // 6,400,000 float4s
    int threads = 256;                            // 8 wave32s per block
    int blocks  = (n4 + threads - 1) / threads;   // 25,000 blocks, no tail

    damping_layer_kernel<<<blocks, threads, 0, stream>>>(input4, kipf4, lbda, out4, n4);
}